// GCN_unc_65223373357462
// MI455X (gfx1250) — compile-verified
//
#include <hip/hip_runtime.h>
#include <hip/hip_bf16.h>

typedef float v2f __attribute__((ext_vector_type(2)));
typedef float v8f __attribute__((ext_vector_type(8)));

#define N_ATOMS   32768
#define N_EDGES   524288
#define IN_DIM    74
#define IN_PAD    128       // pad layer-0 K to 128 so all GEMM K are multiples of 64
#define HID       128
#define NB        4096      // B = N_GRAPHS * NUM_MOLS
#define MAXA      64
#define KC        64        // K-chunk staged in LDS
#define APAD      2         // LDS row pad (floats) to avoid bank conflicts

// ---------------- utility kernels ----------------

__global__ void zero_f32(float* __restrict__ p, long n) {
    long i = (long)blockIdx.x * blockDim.x + threadIdx.x;
    long stride = (long)gridDim.x * blockDim.x;
    for (; i < n; i += stride) p[i] = 0.0f;
}

__global__ void degree_kernel(const int* __restrict__ esrc, const int* __restrict__ edst,
                              float* __restrict__ outdeg, float* __restrict__ indeg, int E) {
    int e = blockIdx.x * blockDim.x + threadIdx.x;
    if (e >= E) return;
    atomicAdd(&outdeg[esrc[e]], 1.0f);
    atomicAdd(&indeg[edst[e]], 1.0f);
}

__global__ void invdeg_kernel(const float* __restrict__ outdeg, const float* __restrict__ indeg,
                              float* __restrict__ inv_out, float* __restrict__ inv_in, int n) {
    int i = blockIdx.x * blockDim.x + threadIdx.x;
    if (i >= n) return;
    inv_out[i] = rsqrtf(fmaxf(outdeg[i], 1.0f));
    inv_in[i]  = rsqrtf(fmaxf(indeg[i], 1.0f));
}

__global__ void padw0_kernel(const float* __restrict__ W0, float* __restrict__ W0p) {
    int i = blockIdx.x * blockDim.x + threadIdx.x;   // IN_PAD*HID threads
    if (i >= IN_PAD * HID) return;
    int r = i >> 7, c = i & 127;
    W0p[i] = (r < IN_DIM) ? W0[r * HID + c] : 0.0f;
}

// scatter: agg[dst][c] += h[src][c] * inv_out[src]   (2 channels per thread)
__global__ void scatter_edges2(const float* __restrict__ h, const float* __restrict__ inv_out,
                               const int* __restrict__ esrc, const int* __restrict__ edst,
                               float* __restrict__ agg, int dim2, int strideH, int strideA,
                               long total) {
    long idx = (long)blockIdx.x * blockDim.x + threadIdx.x;
    if (idx >= total) return;
    int e  = (int)(idx / dim2);
    int c2 = (int)(idx - (long)e * dim2);
    int s = esrc[e], d = edst[e];
    const float2 v = *(const float2*)(h + (size_t)s * strideH + c2 * 2);
    const float sc = inv_out[s];
    float* ap = agg + (size_t)d * strideA + c2 * 2;
    atomicAdd(ap,     v.x * sc);
    atomicAdd(ap + 1, v.y * sc);
}

// ---------------- LDS-tiled WMMA fp32 GEMM: out = act(rowScale.*A @ W + bias) ----------------
// Block: 256 thr = 8 waves. Block tile: 64(M) x 128(N), K chunked by KC=64 through LDS.
// Wave w: rows row0 = bx*64 + (w&3)*16 ; cols col0 = by*128 + (w>>2)*64 (4 WMMA accums).
// Requires M%64==0, N==128*gridDim.y, K%64==0, A stride == K, out stride == N.
__global__ void gemm_bias_act(const float* __restrict__ A, const float* __restrict__ rowScale,
                              const float* __restrict__ W, const float* __restrict__ bias,
                              float* __restrict__ out, int M, int K, int N, int doRelu) {
    // Bs: k-pair interleaved panel  [KC/2][128][2]  -> fragment = one ds_load_b64
    __shared__ float Bs[KC * 128];           // 32 KB
    __shared__ float As[64 * (KC + APAD)];   // 16.5 KB
    const int tid  = threadIdx.x;
    const int lane = tid & 31;
    const int wave = tid >> 5;
    const int rowBlock = blockIdx.x * 64;
    const int colBase  = blockIdx.y * 128;

    const int m    = lane & 15;
    const int kh   = (lane >> 4) * 2;              // 0 or 2: K-subslot of this half-wave
    const int mloc = (wave & 3) * 16 + m;          // local A row for this lane
    const int cloc = (wave >> 2) * 64 + m;         // local B col (first subtile)

    v8f c0 = {}, c1 = {}, c2 = {}, c3 = {};

    for (int kbase = 0; kbase < K; kbase += KC) {
        // ---- stage B chunk: rows kbase..kbase+63, cols colBase..+127 (coalesced b128 loads)
        #pragma unroll
        for (int i = 0; i < (KC * 128) / (256 * 4); ++i) {   // 8 iters
            int j = i * 256 + tid;                 // float4 id
            int k = j >> 5;                        // 32 float4 per row
            int c = (j & 31) * 4;
            float4 v = *(const float4*)(W + (size_t)(kbase + k) * N + colBase + c);
            float* dst = Bs + ((k >> 1) << 8) + (c << 1) + (k & 1);
            dst[0] = v.x; dst[2] = v.y; dst[4] = v.z; dst[6] = v.w;
        }
        // ---- stage A chunk (scaled by rowScale at staging time)
        #pragma unroll
        for (int i = 0; i < (64 * KC) / (256 * 4); ++i) {    // 4 iters
            int j  = i * 256 + tid;                // float4 id
            int r  = j >> 4;                       // 16 float4 per row
            int c4 = (j & 15) * 4;
            float s = (rowScale != nullptr) ? rowScale[rowBlock + r] : 1.0f;
            float4 v = *(const float4*)(A + (size_t)(rowBlock + r) * K + kbase + c4);
            float* dst = As + r * (KC + APAD) + c4;
            dst[0] = v.x * s; dst[1] = v.y * s; dst[2] = v.z * s; dst[3] = v.w * s;
        }
        __syncthreads();

        const float* Ap = As + mloc * (KC + APAD) + kh;
        #pragma unroll
        for (int k0 = 0; k0 < KC; k0 += 4) {
            v2f a = *(const v2f*)(Ap + k0);
            const float* Bp = Bs + (((k0 + kh) >> 1) << 8) + cloc * 2;
            v2f b0 = *(const v2f*)(Bp);
            v2f b1 = *(const v2f*)(Bp + 32);
            v2f b2 = *(const v2f*)(Bp + 64);
            v2f b3 = *(const v2f*)(Bp + 96);
            c0 = __builtin_amdgcn_wmma_f32_16x16x4_f32(false, a, false, b0, (short)0, c0, false, false);
            c1 = __builtin_amdgcn_wmma_f32_16x16x4_f32(false, a, false, b1, (short)0, c1, false, false);
            c2 = __builtin_amdgcn_wmma_f32_16x16x4_f32(false, a, false, b2, (short)0, c2, false, false);
            c3 = __builtin_amdgcn_wmma_f32_16x16x4_f32(false, a, false, b3, (short)0, c3, false, false);
        }
        __syncthreads();
    }

    v8f cc[4] = {c0, c1, c2, c3};
    const int rbase = rowBlock + (wave & 3) * 16 + ((lane >> 4) << 3);  // C[r + 8*(lane>>4)][lane&15]
    const int cbase = colBase + (wave >> 2) * 64;
    #pragma unroll
    for (int t = 0; t < 4; ++t) {
        const int col = cbase + t * 16 + (lane & 15);
        const float bv = bias[col];
        #pragma unroll
        for (int r = 0; r < 8; ++r) {
            float v = cc[t][r] + bv;
            if (doRelu) v = fmaxf(v, 0.0f);
            out[(size_t)(rbase + r) * N + col] = v;
        }
    }
}

// ---------------- small fused kernels ----------------

__global__ void tpart_kernel(const float* __restrict__ T, const float* __restrict__ TW,
                             const float* __restrict__ Tb, float* __restrict__ T_part) {
    int idx = blockIdx.x * blockDim.x + threadIdx.x;  // NB*HID
    int b = idx >> 7, c = idx & 127;
    T_part[idx] = T[b] * TW[c] + Tb[c];
}

// mean_atom[b][c] = (1/64) * sum_a ( h[g][a][c] + relu(u1[g][a][c] + u2[b][c]) ), g = b/8
__global__ void mean_atoms_kernel(const float* __restrict__ h, const float* __restrict__ u1,
                                  const float* __restrict__ u2, float* __restrict__ mean_atom) {
    int b = blockIdx.x;          // 0..NB-1
    int c = threadIdx.x;         // 0..127
    int g = b >> 3;
    const float* __restrict__ hp = h  + (size_t)g * MAXA * HID + c;
    const float* __restrict__ up = u1 + (size_t)g * MAXA * HID + c;
    float t2 = u2[(size_t)b * HID + c];
    float acc = 0.0f;
    for (int a = 0; a < MAXA; ++a)
        acc += hp[a * HID] + fmaxf(up[a * HID] + t2, 0.0f);
    mean_atom[(size_t)b * HID + c] = acc * (1.0f / (float)MAXA);
}

// hr = [ T_part + relu(u3+u4) , mean_atom ]   (NB x 2*HID)
__global__ void build_hr_kernel(const float* __restrict__ T_part, const float* __restrict__ u3,
                                const float* __restrict__ u4, const float* __restrict__ mean_atom,
                                float* __restrict__ hr) {
    int idx = blockIdx.x * blockDim.x + threadIdx.x;  // NB*HID
    int b = idx >> 7, c = idx & 127;
    float ut = T_part[idx] + fmaxf(u3[idx] + u4[idx], 0.0f);
    hr[(size_t)b * 2 * HID + c]       = ut;
    hr[(size_t)b * 2 * HID + HID + c] = mean_atom[idx];
}

// out[b][j] = relu( sum_c r2[b][c] * W[c][j] + bias[j] ),  j in {0,1}
__global__ void readout_final(const float* __restrict__ r2, const float* __restrict__ W,
                              const float* __restrict__ bsc, float* __restrict__ out) {
    int idx = blockIdx.x * blockDim.x + threadIdx.x;  // NB*2
    if (idx >= NB * 2) return;
    int b = idx >> 1, j = idx & 1;
    const float* __restrict__ rp = r2 + (size_t)b * HID;
    float acc = bsc[j];
    for (int c = 0; c < HID; ++c) acc += rp[c] * W[c * 2 + j];
    out[idx] = fmaxf(acc, 0.0f);
}

// ---------------- host orchestration ----------------

extern "C" void kernel_launch(void* const* d_in, const int* in_sizes, int n_in,
                              void* d_out, int out_size, void* d_ws, size_t ws_size,
                              hipStream_t stream) {
    // Input order: setup_inputs() dict insertion order, params nested in insertion order.
    const float* features = (const float*)d_in[0];          // [32768,74]
    const float* T        = (const float*)d_in[1];          // [4096]
    const float* gW[4]    = {(const float*)d_in[2], (const float*)d_in[3],
                             (const float*)d_in[4], (const float*)d_in[5]};
    const float* gB[4]    = {(const float*)d_in[6], (const float*)d_in[7],
                             (const float*)d_in[8], (const float*)d_in[9]};
    const float* dW[8];  const float* dB[8];
    for (int i = 0; i < 8; ++i) { dW[i] = (const float*)d_in[10 + i]; dB[i] = (const float*)d_in[18 + i]; }
    const float* rW[3]    = {(const float*)d_in[26], (const float*)d_in[27], (const float*)d_in[28]};
    const float* rB[3]    = {(const float*)d_in[29], (const float*)d_in[30], (const float*)d_in[31]};
    const float* TembW    = (const float*)d_in[32];         // [1,128]
    const float* TembB    = (const float*)d_in[33];         // [128]
    const int* edge_src   = (const int*)d_in[34];           // [524288]
    const int* edge_dst   = (const int*)d_in[35];
    float* out            = (float*)d_out;                  // [4096,2]

    // workspace layout (floats)
    float* w = (float*)d_ws;
    float* outdeg   = w;                  // 32768
    float* indeg    = outdeg   + N_ATOMS;
    float* inv_out  = indeg    + N_ATOMS;
    float* inv_in   = inv_out  + N_ATOMS;
    float* W0pad    = inv_in   + N_ATOMS;             // 128*128
    float* agg      = W0pad    + IN_PAD * HID;        // 32768*128
    float* h        = agg      + (size_t)N_ATOMS * HID;
    float* u1       = h        + (size_t)N_ATOMS * HID;
    float* u1a      = agg;                            // reuse agg after GCN
    float* T_part   = u1       + (size_t)N_ATOMS * HID;
    float* t2a      = T_part   + (size_t)NB * HID;
    float* u2       = t2a      + (size_t)NB * HID;
    float* mean_a   = u2       + (size_t)NB * HID;
    float* u3a      = mean_a   + (size_t)NB * HID;
    float* u3       = u3a      + (size_t)NB * HID;
    float* u4a      = u3       + (size_t)NB * HID;
    float* u4       = u4a      + (size_t)NB * HID;
    float* hr       = u4       + (size_t)NB * HID;    // NB*256
    float* r1       = hr       + (size_t)NB * 2 * HID;
    float* r2       = r1       + (size_t)NB * 2 * HID;
    (void)ws_size; (void)n_in; (void)in_sizes; (void)out_size;

    const dim3 blk(256);

    // degrees
    zero_f32<<<dim3(1024), blk, 0, stream>>>(outdeg, (long)2 * N_ATOMS);
    degree_kernel<<<dim3(N_EDGES / 256), blk, 0, stream>>>(edge_src, edge_dst, outdeg, indeg, N_EDGES);
    invdeg_kernel<<<dim3(N_ATOMS / 256), blk, 0, stream>>>(outdeg, indeg, inv_out, inv_in, N_ATOMS);
    padw0_kernel<<<dim3((IN_PAD * HID + 255) / 256), blk, 0, stream>>>(gW[0], W0pad);

    // GCN layers
    for (int l = 0; l < 4; ++l) {
        const int dimH   = (l == 0) ? IN_DIM : HID;   // channels scattered
        const float* hin = (l == 0) ? features : h;
        const float* Wm  = (l == 0) ? W0pad : gW[l];
        zero_f32<<<dim3(4096), blk, 0, stream>>>(agg, (long)N_ATOMS * HID);
        const int dim2 = dimH / 2;
        const long total = (long)N_EDGES * dim2;
        scatter_edges2<<<dim3((unsigned)((total + 255) / 256)), blk, 0, stream>>>(
            hin, inv_out, edge_src, edge_dst, agg, dim2, dimH, HID, total);
        gemm_bias_act<<<dim3(N_ATOMS / 64, 1), blk, 0, stream>>>(
            agg, inv_in, Wm, gB[l], h, N_ATOMS, HID, HID, 1);
    }

    // temperature embedding
    tpart_kernel<<<dim3(NB * HID / 256), blk, 0, stream>>>(T, TembW, TembB, T_part);

    // update1 on UNIQUE atoms (8x dedup of the repeated batch)
    gemm_bias_act<<<dim3(N_ATOMS / 64, 1), blk, 0, stream>>>(h,   nullptr, dW[0], dB[0], u1a, N_ATOMS, HID, HID, 0);
    gemm_bias_act<<<dim3(N_ATOMS / 64, 1), blk, 0, stream>>>(u1a, nullptr, dW[1], dB[1], u1,  N_ATOMS, HID, HID, 0);
    // update2 path
    gemm_bias_act<<<dim3(NB / 64, 1), blk, 0, stream>>>(T_part, nullptr, dW[2], dB[2], t2a, NB, HID, HID, 0);
    gemm_bias_act<<<dim3(NB / 64, 1), blk, 0, stream>>>(t2a,    nullptr, dW[3], dB[3], u2,  NB, HID, HID, 0);

    // fused segment-mean: mean_atom = mean_a( h + relu(u1 + u2[b]) )  (ua never materialized)
    mean_atoms_kernel<<<dim3(NB), dim3(HID), 0, stream>>>(h, u1, u2, mean_a);

    // update3 / update4
    gemm_bias_act<<<dim3(NB / 64, 1), blk, 0, stream>>>(mean_a, nullptr, dW[4], dB[4], u3a, NB, HID, HID, 0);
    gemm_bias_act<<<dim3(NB / 64, 1), blk, 0, stream>>>(u3a,    nullptr, dW[5], dB[5], u3,  NB, HID, HID, 0);
    gemm_bias_act<<<dim3(NB / 64, 1), blk, 0, stream>>>(T_part, nullptr, dW[6], dB[6], u4a, NB, HID, HID, 0);
    gemm_bias_act<<<dim3(NB / 64, 1), blk, 0, stream>>>(u4a,    nullptr, dW[7], dB[7], u4,  NB, HID, HID, 0);

    // hr = [T_part + relu(u3+u4), mean_atom]
    build_hr_kernel<<<dim3(NB * HID / 256), blk, 0, stream>>>(T_part, u3, u4, mean_a, hr);

    // readout
    gemm_bias_act<<<dim3(NB / 64, 2), blk, 0, stream>>>(hr, nullptr, rW[0], rB[0], r1, NB, 2 * HID, 2 * HID, 1);
    gemm_bias_act<<<dim3(NB / 64, 1), blk, 0, stream>>>(r1, nullptr, rW[1], rB[1], r2, NB, 2 * HID, HID, 1);
    readout_final<<<dim3((NB * 2 + 255) / 256), blk, 0, stream>>>(r2, rW[2], rB[2], out);
}